// ViewpointLoss_75935021793772
// MI455X (gfx1250) — compile-verified
//
#include <hip/hip_runtime.h>
#include <hip/hip_bf16.h>
#include <math.h>
#include <stdint.h>

#define CP 360
#define NROW 16384
#define W 4320
#define ROWS_PER_BLK 16
#define LDS_STRIDE 364                 // floats; 1456 B rows -> 16B aligned, modest bank spread
#define NBLK (NROW / ROWS_PER_BLK)     // 1024 blocks, one wave32 each

typedef float v2f __attribute__((ext_vector_type(2)));
typedef float v8f __attribute__((ext_vector_type(8)));

__device__ __forceinline__ float pick8(v8f v, int sel) {
  sel &= 7;
  float r = v[0];
  r = (sel == 1) ? v[1] : r;
  r = (sel == 2) ? v[2] : r;
  r = (sel == 3) ? v[3] : r;
  r = (sel == 4) ? v[4] : r;
  r = (sel == 5) ? v[5] : r;
  r = (sel == 6) ? v[6] : r;
  r = (sel == 7) ? v[7] : r;
  return r;
}

__global__ void __launch_bounds__(32)
vp_loss_main(const float* __restrict__ preds,
             const float* __restrict__ labels,
             const int* __restrict__ obj,
             float* __restrict__ ws) {
  __shared__ __align__(16) float sh_p[ROWS_PER_BLK * LDS_STRIDE];
  __shared__ __align__(16) float sh_l[ROWS_PER_BLK * LDS_STRIDE];

  const int lane = threadIdx.x;      // 0..31, wave32
  const int r16  = lane & 15;        // row within tile
  const int row  = blockIdx.x * ROWS_PER_BLK + r16;
  const int cls  = obj[row];
  const unsigned rowoff = ((unsigned)row * (unsigned)W + (unsigned)cls * (unsigned)CP) * 4u;

  const unsigned lds_p0 = (unsigned)(uintptr_t)&sh_p[0];
  const unsigned lds_l0 = (unsigned)(uintptr_t)&sh_l[0];
  const unsigned long long pbase = (unsigned long long)(uintptr_t)preds;
  const unsigned long long lbase = (unsigned long long)(uintptr_t)labels;

  // ---- Stage 16x360 fp32 tiles of preds & labels into LDS via async-to-LDS b128.
  // 90 float4 per row * 16 rows = 1440 transfers per array; 45 iterations, all lanes active.
#pragma unroll 1
  for (int i = lane; i < ROWS_PER_BLK * (CP / 4); i += 32) {
    const int rr = i / (CP / 4);
    const int q  = i - rr * (CP / 4);
    const unsigned go = __shfl(rowoff, rr) + (unsigned)q * 16u;
    const unsigned lo = (unsigned)(rr * (LDS_STRIDE * 4) + q * 16);
    asm volatile("global_load_async_to_lds_b128 %0, %1, %2 offset:0"
                 :: "v"(lds_p0 + lo), "v"(go), "s"(pbase) : "memory");
    asm volatile("global_load_async_to_lds_b128 %0, %1, %2 offset:0"
                 :: "v"(lds_l0 + lo), "v"(go), "s"(lbase) : "memory");
  }
  asm volatile("s_wait_asynccnt 0x0" ::: "memory");

  // ---- WMMA: accD = L(16x360) x P^T(360x16); diag = per-row dot(l,p).
  //      accS = L x Ones       ; any column = per-row sum(l).
  // f32 16x16x4 operand layout: lane L holds M(or N)=L&15, K = 2*(L>>4)+{0,1}.
  // Two accumulators per result (even/odd K-chunks) to deepen the XDL pipeline:
  // RAW distance between dependent wmmas becomes 4 instead of 2.
  v8f accD0 = {}, accD1 = {};
  v8f accS0 = {}, accS1 = {};
  v2f vones;
  vones[0] = 1.0f; vones[1] = 1.0f;
  const int koff = (lane >> 4) * 2;
  const int rbase = r16 * LDS_STRIDE;
  for (int c = 0; c < CP / 4; c += 2) {
    const int k0 = 4 * c;
    v2f a0 = *(const v2f*)&sh_l[rbase + k0 + koff];
    v2f b0 = *(const v2f*)&sh_p[rbase + k0 + koff];
    v2f a1 = *(const v2f*)&sh_l[rbase + k0 + 4 + koff];
    v2f b1 = *(const v2f*)&sh_p[rbase + k0 + 4 + koff];
    accD0 = __builtin_amdgcn_wmma_f32_16x16x4_f32(false, a0, false, b0,
                                                  (short)0, accD0, false, false);
    accS0 = __builtin_amdgcn_wmma_f32_16x16x4_f32(false, a0, false, vones,
                                                  (short)0, accS0, false, false);
    accD1 = __builtin_amdgcn_wmma_f32_16x16x4_f32(false, a1, false, b1,
                                                  (short)0, accD1, false, false);
    accS1 = __builtin_amdgcn_wmma_f32_16x16x4_f32(false, a1, false, vones,
                                                  (short)0, accS1, false, false);
  }
  const v8f accD = accD0 + accD1;
  const v8f accS = accS0 + accS1;

  // Diagonal extraction: C[m][m] lives at (VGPR m, lane m) for m<8 and
  // (VGPR m-8, lane m+16) for m>=8.  Valid source lanes: 0..7 and 24..31.
  const int sel = (lane < 8) ? lane : (lane - 24);
  const float dv = pick8(accD, sel);
  const float sv = pick8(accS, sel);
  const int src = (r16 < 8) ? r16 : (r16 + 16);
  const float dotv  = __shfl(dv, src);   // valid on lanes 0..15: row r16 dot(l,p)
  const float sumlv = __shfl(sv, src);   // valid on lanes 0..15: row r16 sum(l)

  // ---- max + sum(exp) per row: lane r and lane r+16 split row r's 360 floats.
  // Unroll 10 so ds_load_b64s are batched and one dscnt wait covers many loads.
  const int half = lane >> 4;
  const float* prow = &sh_p[rbase + half * (CP / 2)];
  float m = -INFINITY;
#pragma unroll 10
  for (int i = 0; i < CP / 4; ++i) {     // 90 float2 per half-row
    v2f v = *(const v2f*)&prow[2 * i];
    m = fmaxf(m, fmaxf(v[0], v[1]));
  }
  m = fmaxf(m, __shfl_xor(m, 16));       // full-row max
  float s = 0.0f;
#pragma unroll 10
  for (int i = 0; i < CP / 4; ++i) {
    v2f v = *(const v2f*)&prow[2 * i];
    s += __expf(v[0] - m) + __expf(v[1] - m);
  }
  s += __shfl_xor(s, 16);                // full-row sum(exp)
  const float lse = m + __logf(s);

  // ---- per-row partial:  dot(l,p) - lse*sum(l);  reduce 16 rows -> 1 per block.
  float part = (lane < 16) ? (dotv - lse * sumlv) : 0.0f;
  part += __shfl_xor(part, 8);
  part += __shfl_xor(part, 4);
  part += __shfl_xor(part, 2);
  part += __shfl_xor(part, 1);
  if (lane == 0) ws[blockIdx.x] = part;
}

__global__ void __launch_bounds__(256)
vp_finalize(const float* __restrict__ ws, float* __restrict__ out) {
  __shared__ float sm[256];
  const int t = threadIdx.x;
  float s = ws[t] + ws[t + 256] + ws[t + 512] + ws[t + 768];
  sm[t] = s;
  __syncthreads();
  for (int o = 128; o > 0; o >>= 1) {
    if (t < o) sm[t] += sm[t + o];
    __syncthreads();
  }
  if (t == 0) out[0] = -sm[0] * (1.0f / (360.0f * 16384.0f));
}

extern "C" void kernel_launch(void* const* d_in, const int* in_sizes, int n_in,
                              void* d_out, int out_size, void* d_ws, size_t ws_size,
                              hipStream_t stream) {
  const float* preds  = (const float*)d_in[0];
  const float* labels = (const float*)d_in[1];
  const int*   obj    = (const int*)d_in[2];
  float* ws  = (float*)d_ws;   // needs NBLK floats = 4 KB
  float* out = (float*)d_out;

  vp_loss_main<<<NBLK, 32, 0, stream>>>(preds, labels, obj, ws);
  vp_finalize<<<1, 256, 0, stream>>>(ws, out);
}